// Canny_71906342469667
// MI455X (gfx1250) — compile-verified
//
#include <hip/hip_runtime.h>
#include <hip/hip_bf16.h>
#include <stdint.h>

// ---------------------------------------------------------------------------
// Canny (B=8, C=3, H=W=1024) for gfx1250.
//  K1: gray + separable 3x3 Gaussian; column conv done with V_WMMA_F32_16X16X4_F32
//  K2: Sobel + magnitude + NMS + double threshold (LDS tiled)
//  K3: hysteresis pass; tiles loaded with global_load_async_to_lds_b32 (ASYNCcnt)
// ---------------------------------------------------------------------------

#define HH 1024
#define WW 1024
#define NB 8

// Gaussian sigma=1, ksize=3: g = [a,b,a], a=e^-.5/(1+2e^-.5), b=1/(1+2e^-.5)
#define GA 0.2740686178f
#define GB 0.4518627524f

typedef float v2f __attribute__((ext_vector_type(2)));
typedef float v8f __attribute__((ext_vector_type(8)));

__device__ __forceinline__ float wA(int d) {
  // banded column-conv matrix entry: A[m][j] with d = j - m
  return (d == 0 || d == 2) ? GA : (d == 1 ? GB : 0.0f);
}

__device__ __forceinline__ float rowblur(const float* gt, int j, int n) {
  // horizontal blur of gray tile row j (18-wide rows), output col n (0..15)
  if (j >= 18) return 0.0f;
  const float* r = gt + j * 18 + n;
  return GA * r[0] + GB * r[1] + GA * r[2];
}

// -------------------- K1: grayscale + Gaussian blur ------------------------
// 1 wave per 16x16 output tile; 8 waves (8 tiles) per 256-thread block.
// Column blur: D(16x16) = sum_k A_k(16x4) x B_k(4x16), 5 chained WMMAs (K=20).
__global__ void __launch_bounds__(256)
k_gray_blur(const float* __restrict__ img, float* __restrict__ blur) {
  __shared__ float sgray[8][18 * 18];
  const int lane = threadIdx.x & 31;
  const int wid  = threadIdx.x >> 5;
  const int tile = blockIdx.x * 8 + wid;           // 0 .. 8*64*64-1
  const int tcx  = tile & 63;
  const int tcy  = (tile >> 6) & 63;
  const int b    = tile >> 12;

  const float* Rp = img + (size_t)b * 3 * HH * WW;
  const float* Gp = Rp + (size_t)HH * WW;
  const float* Bp = Gp + (size_t)HH * WW;
  const int y0 = tcy * 16 - 1, x0 = tcx * 16 - 1;

  // fill 18x18 gray tile (reflect padding, as in jnp.pad 'reflect')
  float* gt = sgray[wid];
  for (int e = lane; e < 324; e += 32) {
    int yy = y0 + e / 18, xx = x0 + e % 18;
    yy = (yy < 0) ? -yy : (yy >= HH ? 2 * HH - 2 - yy : yy);
    xx = (xx < 0) ? -xx : (xx >= WW ? 2 * WW - 2 - xx : xx);
    size_t p = (size_t)yy * WW + xx;
    gt[e] = 0.299f * Rp[p] + 0.587f * Gp[p] + 0.114f * Bp[p];
  }
  __syncthreads();

  // f32 16x16x4 fragment layout (ISA 7.12.2):
  //  A: lane m=lane&15; VGPR0 = K=(2*half), VGPR1 = K=(2*half+1)
  //  B: lane n=lane&15; VGPR0 = row (2*half), VGPR1 = row (2*half+1)
  //  D: lane n=lane&15; VGPR r = (M = r + 8*half, N = n)
  const int nn   = lane & 15;
  const int half = lane >> 4;
  v8f acc = {};
#pragma unroll
  for (int k = 0; k < 5; ++k) {
    const int j0 = 4 * k + 2 * half;
    v2f a, bv;
    a.x  = wA(j0 - nn);
    a.y  = wA(j0 + 1 - nn);
    bv.x = rowblur(gt, j0, nn);
    bv.y = rowblur(gt, j0 + 1, nn);
    acc = __builtin_amdgcn_wmma_f32_16x16x4_f32(
        /*neg_a=*/false, a, /*neg_b=*/false, bv,
        /*c_mod=*/(short)0, acc, /*reuse_a=*/false, /*reuse_b=*/false);
  }

  float* out = blur + (size_t)b * HH * WW;
  const int col = tcx * 16 + nn;
#pragma unroll
  for (int r = 0; r < 8; ++r) {
    const int row = tcy * 16 + r + 8 * half;
    out[(size_t)row * WW + col] = acc[r];
  }
}

// -------------------- K2: Sobel + magnitude + NMS + threshold --------------
__global__ void __launch_bounds__(256)
k_sobel_nms(const float* __restrict__ blur, float* __restrict__ mag_out,
            float* __restrict__ edge_out) {
  __shared__ float bl[20][21];   // blurred, halo 2, edge-clamped
  __shared__ float mg[18][19];   // magnitude, halo 1
  const int b  = blockIdx.z;
  const int tx = threadIdx.x, ty = threadIdx.y;
  const int y0 = blockIdx.y * 16, x0 = blockIdx.x * 16;
  const float* bp = blur + (size_t)b * HH * WW;
  const int t = ty * 16 + tx;

  for (int e = t; e < 400; e += 256) {
    int yy = y0 + e / 20 - 2, xx = x0 + e % 20 - 2;
    yy = min(max(yy, 0), HH - 1);
    xx = min(max(xx, 0), WW - 1);
    bl[e / 20][e % 20] = bp[(size_t)yy * WW + xx];
  }
  __syncthreads();

  for (int e = t; e < 324; e += 256) {
    const int jy = e / 18, jx = e % 18;   // window base in bl
    float gx = -bl[jy][jx] + bl[jy][jx + 2]
               - 2.f * bl[jy + 1][jx] + 2.f * bl[jy + 1][jx + 2]
               - bl[jy + 2][jx] + bl[jy + 2][jx + 2];
    float gy = -bl[jy][jx] - 2.f * bl[jy][jx + 1] - bl[jy][jx + 2]
               + bl[jy + 2][jx] + 2.f * bl[jy + 2][jx + 1] + bl[jy + 2][jx + 2];
    mg[jy][jx] = sqrtf(gx * gx + gy * gy + 1e-6f);
  }
  __syncthreads();

  // center pixel gradient for direction
  const int jy = ty + 1, jx = tx + 1;
  float gx = -bl[jy][jx] + bl[jy][jx + 2]
             - 2.f * bl[jy + 1][jx] + 2.f * bl[jy + 1][jx + 2]
             - bl[jy + 2][jx] + bl[jy + 2][jx + 2];
  float gy = -bl[jy][jx] - 2.f * bl[jy][jx + 1] - bl[jy][jx + 2]
             + bl[jy + 2][jx] + 2.f * bl[jy + 2][jx + 1] + bl[jy + 2][jx + 2];
  const float m = mg[ty + 1][tx + 1];

  // direction = round(angle/45deg), RNE like jnp.round
  const float ang = atan2f(gy, gx) * 1.27323954473516f;  // 4/pi
  int d = (int)rintf(ang);                               // -4..4
  const int pos = (d + 8) & 7;
  const int neg = (pos + 4) & 7;
  const int dy8[8] = {0, 1, 1, 1, 0, -1, -1, -1};
  const int dx8[8] = {1, 1, 0, -1, -1, -1, 0, 1};

  const int yg = y0 + ty, xg = x0 + tx;
  float npos = 0.f, nneg = 0.f;
  {
    int ny = yg + dy8[pos], nx = xg + dx8[pos];
    if (ny >= 0 && ny < HH && nx >= 0 && nx < WW)
      npos = mg[ty + 1 + dy8[pos]][tx + 1 + dx8[pos]];
    ny = yg + dy8[neg]; nx = xg + dx8[neg];
    if (ny >= 0 && ny < HH && nx >= 0 && nx < WW)
      nneg = mg[ty + 1 + dy8[neg]][tx + 1 + dx8[neg]];
  }
  const bool ismax = (m - npos > 0.f) && (m - nneg > 0.f);
  const float mo = ismax ? m : 0.f;
  const float ev = (mo > 0.1f ? 0.5f : 0.f) + (mo > 0.2f ? 0.5f : 0.f);

  const size_t p = (size_t)b * HH * WW + (size_t)yg * WW + xg;
  mag_out[p]  = mo;
  edge_out[p] = ev;
}

// -------------------- K3: hysteresis pass (async LDS tiles) ----------------
__device__ __forceinline__ unsigned to_lds_off(const float* p) {
  return (unsigned)(uintptr_t)(__attribute__((address_space(3))) const float*)p;
}

__global__ void __launch_bounds__(256)
k_hyst(float* __restrict__ edges) {
  __shared__ float et[18][18];
  const int b  = blockIdx.z;
  const int tx = threadIdx.x, ty = threadIdx.y;
  const int t  = ty * 16 + tx;
  const int y0 = blockIdx.y * 16 - 1, x0 = blockIdx.x * 16 - 1;
  float* ep = edges + (size_t)b * HH * WW;

  // DMA the in-image tile entries straight into LDS; zero the out-of-image halo
  for (int i = t; i < 324; i += 256) {
    const int iy = i / 18, ix = i % 18;
    const int yy = y0 + iy, xx = x0 + ix;
    if (yy >= 0 && yy < HH && xx >= 0 && xx < WW) {
      const unsigned lds = to_lds_off(&et[iy][ix]);
      const float* g = ep + (size_t)yy * WW + xx;
      asm volatile("global_load_async_to_lds_b32 %0, %1, off"
                   :: "v"(lds), "v"(g) : "memory");
    } else {
      et[iy][ix] = 0.f;
    }
  }
  asm volatile("s_wait_asynccnt 0x0" ::: "memory");
  __syncthreads();

  // monotone weak->strong promotion, 16 in-tile iterations
  for (int it = 0; it < 16; ++it) {
    const int cy = ty + 1, cx = tx + 1;
    float c = et[cy][cx];
    const bool weak = (c > 0.25f) && (c < 0.75f);
    bool promote = false;
    if (weak) {
      float mx = fmaxf(fmaxf(fmaxf(et[cy - 1][cx - 1], et[cy - 1][cx]),
                             fmaxf(et[cy - 1][cx + 1], et[cy][cx - 1])),
                       fmaxf(fmaxf(et[cy][cx + 1], et[cy + 1][cx - 1]),
                             fmaxf(et[cy + 1][cx], et[cy + 1][cx + 1])));
      promote = (mx > 0.75f);
    }
    __syncthreads();
    if (promote) et[cy][cx] = 1.0f;
    __syncthreads();
  }

  ep[(size_t)(y0 + 1 + ty) * WW + (x0 + 1 + tx)] = et[ty + 1][tx + 1];
}

// ---------------------------------------------------------------------------
extern "C" void kernel_launch(void* const* d_in, const int* in_sizes, int n_in,
                              void* d_out, int out_size, void* d_ws, size_t ws_size,
                              hipStream_t stream) {
  const float* img = (const float*)d_in[0];
  float* out = (float*)d_out;
  float* mag = out;                                 // 8*1024*1024 floats
  float* edg = out + (size_t)NB * HH * WW;          // 8*1024*1024 floats
  float* blr = (float*)d_ws;                        // 32 MB scratch

  // K1: 8*64*64 = 32768 tiles, 8 tiles (waves) per block
  k_gray_blur<<<4096, 256, 0, stream>>>(img, blr);

  dim3 grd(WW / 16, HH / 16, NB), blk(16, 16);
  k_sobel_nms<<<grd, blk, 0, stream>>>(blr, mag, edg);

  for (int p = 0; p < 16; ++p)
    k_hyst<<<grd, blk, 0, stream>>>(edg);
}